// NodePredictor_75041668596277
// MI455X (gfx1250) — compile-verified
//
#include <hip/hip_runtime.h>
#include <hip/hip_bf16.h>
#include <math.h>

typedef float v2f __attribute__((ext_vector_type(2)));
typedef float v8f __attribute__((ext_vector_type(8)));

#define HID 32
#define FEAT3 96
#define EPS_NORM 1e-12f

// ---------------------------------------------------------------------------
// Degree kernels
// ---------------------------------------------------------------------------
__global__ void deg_init_kernel(float* __restrict__ deg, int n) {
    int i = blockIdx.x * blockDim.x + threadIdx.x;
    if (i < n) deg[i] = 1.0f;  // self-loop weight = 1
}

__global__ void deg_scatter_kernel(const int* __restrict__ dst,
                                   const float* __restrict__ ew,
                                   float* __restrict__ deg, int nedges) {
    int e = blockIdx.x * blockDim.x + threadIdx.x;
    if (e < nedges) atomicAdd(&deg[dst[e]], ew[e]);
}

__global__ void deg_finalize_kernel(float* __restrict__ deg, int n) {
    int i = blockIdx.x * blockDim.x + threadIdx.x;
    if (i < n) {
        float d = deg[i];
        deg[i] = (d > 0.0f) ? rsqrtf(d) : 0.0f;  // in place: deg -> dinv
    }
}

// ---------------------------------------------------------------------------
// WMMA f32 GEMM:  H[nrows,32] = X[nrows,K] @ W[K,32]
// Block = 128 threads (4 waves). Block tile = 64 rows x 32 cols.
// Each wave: one 16-row stripe, two 16x16 accumulators, K stepped by 4 via
// V_WMMA_F32_16X16X4_F32.
// A layout  (16x4 f32): lane m = lane&15 holds row M=m; lane>>4 selects K pair;
//                       vgpr {0,1} = K {2*(lane>>4), +1}.
// C/D layout: lane n = lane&15 is column; vgpr v is row v + 8*(lane>>4).
// ---------------------------------------------------------------------------
__global__ __launch_bounds__(128) void gemm32_wmma_kernel(
    const float* __restrict__ X, int ldx, int K,
    const float* __restrict__ W,   // [K,32] row-major
    float* __restrict__ H,         // [nrows,32]
    int nrows) {
    __shared__ float sW[128 * HID];   // up to K=128
    __shared__ float sX[64 * 128];    // 64-row tile, up to K=128

    const int tid  = threadIdx.x;
    const int base = blockIdx.x * 64;

    for (int i = tid; i < K * HID; i += 128) sW[i] = W[i];

    const int total = 64 * K;
    for (int i = tid; i < total; i += 128) {
        int r = i / K, c = i - r * K;
        int row = base + r;
        sX[i] = (row < nrows) ? X[(long)row * ldx + c] : 0.0f;
    }
    __syncthreads();

    const int wv   = tid >> 5;
    const int lane = tid & 31;
    const int m    = lane & 15;   // A: row within stripe; B/D: column
    const int kh   = lane >> 4;   // K-pair select / D row-half select

    v8f acc0 = {};  // columns 0..15
    v8f acc1 = {};  // columns 16..31
    const float* xrow = &sX[(wv * 16 + m) * K];

    for (int k0 = 0; k0 < K; k0 += 4) {
        const int kk = k0 + 2 * kh;
        v2f a;  a.x  = xrow[kk];                 a.y  = xrow[kk + 1];
        v2f b0; b0.x = sW[kk * HID + m];         b0.y = sW[(kk + 1) * HID + m];
        v2f b1; b1.x = sW[kk * HID + 16 + m];    b1.y = sW[(kk + 1) * HID + 16 + m];
        acc0 = __builtin_amdgcn_wmma_f32_16x16x4_f32(false, a, false, b0,
                                                     (short)0, acc0, false, false);
        acc1 = __builtin_amdgcn_wmma_f32_16x16x4_f32(false, a, false, b1,
                                                     (short)0, acc1, false, false);
    }

    #pragma unroll
    for (int v = 0; v < 8; ++v) {
        int row = base + wv * 16 + v + 8 * kh;
        if (row < nrows) {
            H[(long)row * HID + m]      = acc0[v];
            H[(long)row * HID + 16 + m] = acc1[v];
        }
    }
}

// ---------------------------------------------------------------------------
// agg[n,f] = b[f] + h[n,f] * dinv[n]^2        (self-loop term + bias)
// ---------------------------------------------------------------------------
__global__ void init_agg_kernel(const float* __restrict__ h,
                                const float* __restrict__ dinv,
                                const float* __restrict__ b,
                                float* __restrict__ agg, int nrows) {
    long i = (long)blockIdx.x * blockDim.x + threadIdx.x;
    if (i >= (long)nrows * HID) return;
    int n = (int)(i >> 5);
    int f = (int)(i & 31);
    float di = dinv[n];
    agg[i] = b[f] + h[i] * di * di;
}

// ---------------------------------------------------------------------------
// One wave32 per edge; lane = feature. Coalesced 128B gather of h[src,*],
// coalesced 128B atomic-add burst into agg[dst,*].
// ---------------------------------------------------------------------------
__global__ __launch_bounds__(256) void edge_scatter_kernel(
    const int* __restrict__ src, const int* __restrict__ dst,
    const float* __restrict__ ew, const float* __restrict__ dinv,
    const float* __restrict__ h, float* __restrict__ agg, int nedges) {
    int e    = (int)((blockIdx.x * blockDim.x + threadIdx.x) >> 5);
    int lane = threadIdx.x & 31;
    if (e >= nedges) return;
    int   s  = src[e];
    int   d  = dst[e];
    float nm = dinv[s] * ew[e] * dinv[d];
    float v  = h[(long)s * HID + lane] * nm;
    atomicAdd(&agg[(long)d * HID + lane], v);
}

// ---------------------------------------------------------------------------
// Row L2-normalize + ReLU; write into interleaved feats[n, 32*layer + f]
// ---------------------------------------------------------------------------
__global__ __launch_bounds__(256) void norm_relu_kernel(
    const float* __restrict__ agg, float* __restrict__ feats,
    int layer, int nrows) {
    int n    = (int)((blockIdx.x * blockDim.x + threadIdx.x) >> 5);
    int lane = threadIdx.x & 31;
    if (n >= nrows) return;
    float v  = agg[(long)n * HID + lane];
    float ss = v * v;
    #pragma unroll
    for (int m = 16; m >= 1; m >>= 1) ss += __shfl_xor(ss, m, 32);
    float nrm = sqrtf(ss);
    float o   = v / fmaxf(nrm, EPS_NORM);
    o = fmaxf(o, 0.0f);
    feats[(long)n * FEAT3 + layer * HID + lane] = o;
}

// ---------------------------------------------------------------------------
// out[n,10] = feats[n,96] @ W_lin[96,10] + b_lin
// ---------------------------------------------------------------------------
__global__ __launch_bounds__(128) void final_linear_kernel(
    const float* __restrict__ feats, const float* __restrict__ Wl,
    const float* __restrict__ bl, float* __restrict__ out, int nrows) {
    __shared__ float sW[FEAT3 * 10];
    __shared__ float sb[10];
    __shared__ float sF[128 * FEAT3];

    const int tid  = threadIdx.x;
    const int base = blockIdx.x * 128;

    for (int i = tid; i < FEAT3 * 10; i += 128) sW[i] = Wl[i];
    if (tid < 10) sb[tid] = bl[tid];
    for (int i = tid; i < 128 * FEAT3; i += 128) {
        int r = i / FEAT3, c = i - r * FEAT3;
        int row = base + r;
        sF[i] = (row < nrows) ? feats[(long)row * FEAT3 + c] : 0.0f;
    }
    __syncthreads();

    int row = base + tid;
    if (row >= nrows) return;
    const float* f = &sF[tid * FEAT3];
    float acc[10];
    #pragma unroll
    for (int c = 0; c < 10; ++c) acc[c] = sb[c];
    for (int k = 0; k < FEAT3; ++k) {
        float fv = f[k];
        #pragma unroll
        for (int c = 0; c < 10; ++c) acc[c] += fv * sW[k * 10 + c];
    }
    #pragma unroll
    for (int c = 0; c < 10; ++c) out[(long)row * 10 + c] = acc[c];
}

// ---------------------------------------------------------------------------
// Launch
// ---------------------------------------------------------------------------
extern "C" void kernel_launch(void* const* d_in, const int* in_sizes, int n_in,
                              void* d_out, int out_size, void* d_ws, size_t ws_size,
                              hipStream_t stream) {
    const float* x    = (const float*)d_in[0];
    const int*   eidx = (const int*)d_in[1];
    const float* ew   = (const float*)d_in[2];
    const float* W1   = (const float*)d_in[3];
    const float* b1   = (const float*)d_in[4];
    const float* W2   = (const float*)d_in[5];
    const float* b2   = (const float*)d_in[6];
    const float* W3   = (const float*)d_in[7];
    const float* b3   = (const float*)d_in[8];
    const float* Wl   = (const float*)d_in[9];
    const float* bl   = (const float*)d_in[10];

    const int N = in_sizes[0] / 128;   // 100000
    const int E = in_sizes[2];         // 3200000
    const int* src = eidx;
    const int* dst = eidx + E;

    // Workspace layout: dinv[N] | h[N*32] | agg[N*32] | feats[N*96]
    float* ws    = (float*)d_ws;
    float* dinv  = ws;
    float* h     = dinv + N;
    float* agg   = h + (size_t)N * HID;
    float* feats = agg + (size_t)N * HID;
    float* out   = (float*)d_out;

    const int T = 256;
    // degree -> dinv
    deg_init_kernel<<<(N + T - 1) / T, T, 0, stream>>>(dinv, N);
    deg_scatter_kernel<<<(E + T - 1) / T, T, 0, stream>>>(dst, ew, dinv, E);
    deg_finalize_kernel<<<(N + T - 1) / T, T, 0, stream>>>(dinv, N);

    const float* Ws[3] = {W1, W2, W3};
    const float* bs[3] = {b1, b2, b3};

    long nf    = (long)N * HID;
    int  nfblk = (int)((nf + T - 1) / T);
    int  eblk  = (int)(((long)E * 32 + T - 1) / T);

    for (int l = 0; l < 3; ++l) {
        const float* xin = (l == 0) ? x : (feats + HID * (l - 1));
        const int ldx    = (l == 0) ? 128 : FEAT3;
        const int K      = (l == 0) ? 128 : HID;

        gemm32_wmma_kernel<<<(N + 63) / 64, 128, 0, stream>>>(xin, ldx, K, Ws[l], h, N);
        init_agg_kernel<<<nfblk, T, 0, stream>>>(h, dinv, bs[l], agg, N);
        edge_scatter_kernel<<<eblk, T, 0, stream>>>(src, dst, ew, dinv, h, agg, E);
        norm_relu_kernel<<<nfblk, T, 0, stream>>>(agg, feats, l, N);
    }

    final_linear_kernel<<<(N + 127) / 128, 128, 0, stream>>>(feats, Wl, bl, out, N);
}